// Net_16226386444406
// MI455X (gfx1250) — compile-verified
//
#include <hip/hip_runtime.h>
#include <hip/hip_bf16.h>

typedef __attribute__((ext_vector_type(2))) float v2f;
typedef __attribute__((ext_vector_type(8))) float v8f;

// ---------------------------------------------------------------------------
// h = X @ W^T for X:[nrows,64], W:[64,64] row-major (W[o][k]); one wave per
// 16x16 output tile, K accumulated in 16 steps of V_WMMA_F32_16X16X4_F32.
// Requires nrows % 32 == 0 (N = 100000 qualifies); EXEC stays all-ones.
// ---------------------------------------------------------------------------
__global__ void gemm_wmma_64(const float* __restrict__ X,
                             const float* __restrict__ W,
                             float* __restrict__ Hout, int nrows) {
  const int lane = threadIdx.x & 31;
  const int wv   = threadIdx.x >> 5;                 // 0..7 (8 waves / block)
  const int row_base = blockIdx.x * 32 + (wv >> 2) * 16;
  const int col_base = (wv & 3) * 16;
  if (row_base >= nrows) return;                     // wave-uniform

  const int m    = lane & 15;                        // M (A) / N (B) index
  const int koff = (lane >> 4) * 2;                  // K sub-offset per half

  // A[m][k] = X[row_base+m][k]      -> lanes 0-15: K=k,k+1; 16-31: K=k+2,k+3
  // B[k][n] = W[col_base+n][k]      -> identical indexing on W rows
  const float* xrow = X + (size_t)(row_base + m) * 64 + koff;
  const float* wrow = W + (size_t)(col_base + m) * 64 + koff;

  v8f c = {0.f, 0.f, 0.f, 0.f, 0.f, 0.f, 0.f, 0.f};
#pragma unroll
  for (int k = 0; k < 64; k += 4) {
    v2f a = *(const v2f*)(xrow + k);
    v2f b = *(const v2f*)(wrow + k);
    c = __builtin_amdgcn_wmma_f32_16x16x4_f32(false, a, false, b,
                                              (short)0, c, false, false);
  }

  // C layout: c[i] -> row (i + 8*half), col = lane&15
  const int mhi = (lane >> 4) * 8;
  float* orow = Hout + (size_t)row_base * 64 + col_base + m;
#pragma unroll
  for (int i = 0; i < 8; ++i)
    orow[(size_t)(i + mhi) * 64] = c[i];
}

// ---------------------------------------------------------------------------
__global__ void zero_f32(float* __restrict__ p, long long n) {
  long long i = (long long)blockIdx.x * blockDim.x + threadIdx.x;
  if (i < n) p[i] = 0.f;
}

__global__ void deg_edges(const int* __restrict__ dst, const float* __restrict__ w,
                          float* __restrict__ deg, int E) {
  int e = blockIdx.x * blockDim.x + threadIdx.x;
  if (e < E) atomicAdd(&deg[dst[e]], w[e]);
}

__global__ void deg_to_dinv(float* __restrict__ degdinv, int n) {
  int i = blockIdx.x * blockDim.x + threadIdx.x;
  if (i < n) {
    float d = degdinv[i] + 1.0f;                     // + self-loop weight 1
    degdinv[i] = (d > 0.f) ? rsqrtf(d) : 0.f;
  }
}

// 16 threads per edge, float4 gather from h[src], fp32 atomic scatter to agg[dst]
__global__ void agg_edges(const float* __restrict__ Hm,
                          const int* __restrict__ src, const int* __restrict__ dst,
                          const float* __restrict__ w, const float* __restrict__ dinv,
                          float* __restrict__ agg, int E) {
  long long t = (long long)blockIdx.x * blockDim.x + threadIdx.x;
  int e = (int)(t >> 4);
  if (e >= E) return;
  int part = (int)(t & 15);
  int s = src[e], d = dst[e];
  float nrm = dinv[s] * w[e] * dinv[d];
  const float4 hv = *(const float4*)(Hm + (size_t)s * 64 + part * 4);
  float* ap = agg + (size_t)d * 64 + part * 4;
  atomicAdd(ap + 0, nrm * hv.x);
  atomicAdd(ap + 1, nrm * hv.y);
  atomicAdd(ap + 2, nrm * hv.z);
  atomicAdd(ap + 3, nrm * hv.w);
}

// self-loop + bias + ReLU, write out, block-reduce BN statistics
__global__ void finish_stats(const float* __restrict__ agg, const float* __restrict__ Hm,
                             const float* __restrict__ dinv, const float* __restrict__ bias,
                             float* __restrict__ outp, float* __restrict__ sum,
                             float* __restrict__ sumsq, int n) {
  __shared__ float s1[256];
  __shared__ float s2[256];
  const int f = threadIdx.x & 63;
  const int g = threadIdx.x >> 6;                    // 0..3
  const float bf = bias[f];
  float ls = 0.f, lq = 0.f;
  const int base = blockIdx.x * 64;
  for (int j = g; j < 64; j += 4) {
    int node = base + j;
    if (node < n) {
      float di = dinv[node];
      float v = agg[(size_t)node * 64 + f] + di * di * Hm[(size_t)node * 64 + f] + bf;
      v = (v > 0.f) ? v : 0.f;                       // ReLU
      outp[(size_t)node * 64 + f] = v;
      ls += v;
      lq += v * v;
    }
  }
  s1[threadIdx.x] = ls;
  s2[threadIdx.x] = lq;
  __syncthreads();
  if (g == 0) {
    ls = s1[f] + s1[f + 64] + s1[f + 128] + s1[f + 192];
    lq = s2[f] + s2[f + 64] + s2[f + 128] + s2[f + 192];
    atomicAdd(&sum[f], ls);
    atomicAdd(&sumsq[f], lq);
  }
}

__global__ void bn_finalize(const float* __restrict__ sum, const float* __restrict__ sumsq,
                            const float* __restrict__ gam, const float* __restrict__ bet,
                            float* __restrict__ scale, float* __restrict__ shift, float invn) {
  int f = threadIdx.x;                               // 64 threads
  float m  = sum[f] * invn;
  float v  = sumsq[f] * invn - m * m;                // biased variance
  float sc = gam[f] * rsqrtf(v + 1e-5f);
  scale[f] = sc;
  shift[f] = bet[f] - m * sc;
}

__global__ void bn_apply(float* __restrict__ outp, const float* __restrict__ scale,
                         const float* __restrict__ shift, long long n4) {
  long long i = (long long)blockIdx.x * blockDim.x + threadIdx.x;
  if (i >= n4) return;
  int fq = (int)(i & 15) * 4;                        // feature base (64/4 = 16)
  float4 v = ((float4*)outp)[i];
  const float4 sc = *(const float4*)(scale + fq);
  const float4 sh = *(const float4*)(shift + fq);
  v.x = v.x * sc.x + sh.x;
  v.y = v.y * sc.y + sh.y;
  v.z = v.z * sc.z + sh.z;
  v.w = v.w * sc.w + sh.w;
  ((float4*)outp)[i] = v;
}

// out[n] = concat(o1,o2,o3)[n] . Wl + bl ; one wave32 per node
__global__ void final_linear(const float* __restrict__ o1, const float* __restrict__ o2,
                             const float* __restrict__ o3, const float* __restrict__ Wl,
                             const float* __restrict__ bl, float* __restrict__ out, int n) {
  int lane = threadIdx.x & 31;
  int node = blockIdx.x * 8 + (threadIdx.x >> 5);
  if (node >= n) return;                             // wave-uniform
  size_t off = (size_t)node * 64 + lane * 2;
  int j = lane * 2;
  float2 a = *(const float2*)(o1 + off);
  float2 b = *(const float2*)(o2 + off);
  float2 c = *(const float2*)(o3 + off);
  float acc = a.x * Wl[j]       + a.y * Wl[j + 1]
            + b.x * Wl[64 + j]  + b.y * Wl[64 + j + 1]
            + c.x * Wl[128 + j] + c.y * Wl[128 + j + 1];
#pragma unroll
  for (int o = 16; o > 0; o >>= 1) acc += __shfl_down(acc, o, 32);
  if (lane == 0) out[node] = acc + bl[0];
}

// ---------------------------------------------------------------------------
extern "C" void kernel_launch(void* const* d_in, const int* in_sizes, int n_in,
                              void* d_out, int out_size, void* d_ws, size_t ws_size,
                              hipStream_t stream) {
  const int N = in_sizes[0] / 64;
  const int E = in_sizes[1] / 2;

  const float* x   = (const float*)d_in[0];
  const int*   ei  = (const int*)d_in[1];            // int32 (JAX x64 disabled)
  const int*   src = ei;
  const int*   dst = ei + E;
  const float* wgt = (const float*)d_in[2];

  const float* Wm[3] = { (const float*)d_in[3],  (const float*)d_in[7],  (const float*)d_in[11] };
  const float* bs[3] = { (const float*)d_in[4],  (const float*)d_in[8],  (const float*)d_in[12] };
  const float* gs[3] = { (const float*)d_in[5],  (const float*)d_in[9],  (const float*)d_in[13] };
  const float* es[3] = { (const float*)d_in[6],  (const float*)d_in[10], (const float*)d_in[14] };
  const float* Wl = (const float*)d_in[15];
  const float* bl = (const float*)d_in[16];

  // workspace layout (floats)
  float* ws      = (float*)d_ws;
  float* degdinv = ws;                               // N
  float* Hbuf    = degdinv + N;                      // N*64
  float* AGG     = Hbuf + (size_t)N * 64;            // N*64
  float* stats   = AGG + (size_t)N * 64;             // 256: sum,sumsq,scale,shift
  float* sum     = stats;
  float* sumsq   = stats + 64;
  float* scale   = stats + 128;
  float* shift   = stats + 192;
  float* OUT[3];
  OUT[0] = stats + 256;
  OUT[1] = OUT[0] + (size_t)N * 64;
  OUT[2] = OUT[1] + (size_t)N * 64;

  const int TB = 256;
  // ---- degrees -> dinv (graph-constant, shared by all 3 layers) ----
  zero_f32<<<(N + TB - 1) / TB, TB, 0, stream>>>(degdinv, N);
  deg_edges<<<(E + TB - 1) / TB, TB, 0, stream>>>(dst, wgt, degdinv, E);
  deg_to_dinv<<<(N + TB - 1) / TB, TB, 0, stream>>>(degdinv, N);

  const long long aggN   = (long long)N * 64 + 128;  // AGG + sum + sumsq
  const long long edgeT  = (long long)E * 16;
  const long long n4     = (long long)N * 16;

  const float* A = x;
  for (int l = 0; l < 3; ++l) {
    gemm_wmma_64<<<N / 32, TB, 0, stream>>>(A, Wm[l], Hbuf, N);
    zero_f32<<<(int)((aggN + TB - 1) / TB), TB, 0, stream>>>(AGG, aggN);
    agg_edges<<<(int)((edgeT + TB - 1) / TB), TB, 0, stream>>>(Hbuf, src, dst, wgt,
                                                               degdinv, AGG, E);
    finish_stats<<<(N + 63) / 64, TB, 0, stream>>>(AGG, Hbuf, degdinv, bs[l],
                                                   OUT[l], sum, sumsq, N);
    bn_finalize<<<1, 64, 0, stream>>>(sum, sumsq, gs[l], es[l], scale, shift, 1.0f / N);
    bn_apply<<<(int)((n4 + TB - 1) / TB), TB, 0, stream>>>(OUT[l], scale, shift, n4);
    A = OUT[l];
  }

  final_linear<<<(N + 7) / 8, TB, 0, stream>>>(OUT[0], OUT[1], OUT[2], Wl, bl,
                                               (float*)d_out, N);
}